// Attention_37915971289495
// MI455X (gfx1250) — compile-verified
//
#include <hip/hip_runtime.h>
#include <math.h>

#define LL    2048
#define DDIM  1024
#define NH    16
#define DHD   64
#define SCALE_ 0.125f
#define NEGV  (-1000000000.0f)

typedef __attribute__((ext_vector_type(2))) float v2f;
typedef __attribute__((ext_vector_type(8))) float v8f;

// LDS layout (per buffer): As[128][36] (pad 36: multiple of 4 for aligned
// B128 stores, stride gives conflict-free 16-lane column gathers),
// Bs[32][68].
#define AS_STRIDE 36
#define BS_STRIDE 68
#define AS_FLOATS (128 * AS_STRIDE)            // 4608
#define BS_FLOATS (32 * BS_STRIDE)             // 2176
#define BUF_FLOATS (AS_FLOATS + BS_FLOATS)     // 6784
#define SMEM_FLOATS (2 * BUF_FLOATS)           // 13568 (~53 KB)

// ------------------------------------------------------------------
// 128x64-tile fp32 WMMA GEMM engine, 4 waves / 128 threads.
// Wave w computes rows [tileM+32w, +32) x cols [tileN, +64):
// acc[rt][nt] = 16x16 tile, rt in {0,1} row halves, nt in {0..3} col tiles.
// Double-buffered LDS staging of K-chunks of 32. The __shared__ array is
// declared HERE and indexed directly with integer offsets so every access
// stays addrspace(3) -> ds_load/ds_store (no generic/flat fallback).
// A: [M x K] row-major (lda). B: [K x N] row-major (ldb), or if TRANSB,
// B is [N x K] row-major and we compute A * B^T.
// ------------------------------------------------------------------
template<bool TRANSB>
__device__ __forceinline__ void gemm_tile_128x64(
    const float* __restrict__ A, const float* __restrict__ B,
    int lda, int ldb, int K, int tileM, int tileN,
    v8f acc[2][4])
{
  __shared__ float smem[SMEM_FLOATS];

  const int tid   = threadIdx.x;
  const int wave  = tid >> 5;
  const int lane  = tid & 31;
  const int l15   = lane & 15;
  const int lhalf = lane >> 4;     // 0: K pair {0,1}, 1: K pair {2,3} (ISA layout)

  auto stage = [&](int buf, int kk) {
    const int aoff = buf * BUF_FLOATS;
    const int boff = buf * BUF_FLOATS + AS_FLOATS;
    // A chunk 128x32, float4-vectorized (coalesced along k)
    for (int i = tid; i < 128 * 8; i += 128) {
      int r = i >> 3, c4 = i & 7;
      float4 vv = *(const float4*)&A[(size_t)(tileM + r) * lda + kk + (c4 << 2)];
      *(float4*)&smem[aoff + r * AS_STRIDE + (c4 << 2)] = vv;
    }
    if (!TRANSB) {
      // B chunk 32x64 as Bs[k][n], float4 along n (coalesced)
      for (int i = tid; i < 32 * 16; i += 128) {
        int r = i >> 4, c4 = i & 15;
        float4 vv = *(const float4*)&B[(size_t)(kk + r) * ldb + tileN + (c4 << 2)];
        *(float4*)&smem[boff + r * BS_STRIDE + (c4 << 2)] = vv;
      }
    } else {
      // B is [N x K]: float4 reads along k (coalesced), transpose into Bs[k][n]
      for (int i = tid; i < 64 * 8; i += 128) {
        int n = i >> 3, k4 = i & 7;
        float4 vv = *(const float4*)&B[(size_t)(tileN + n) * ldb + kk + (k4 << 2)];
        int kb = (k4 << 2);
        smem[boff + (kb + 0) * BS_STRIDE + n] = vv.x;
        smem[boff + (kb + 1) * BS_STRIDE + n] = vv.y;
        smem[boff + (kb + 2) * BS_STRIDE + n] = vv.z;
        smem[boff + (kb + 3) * BS_STRIDE + n] = vv.w;
      }
    }
  };

  stage(0, 0);
  int cur = 0;
  for (int kk = 0; kk < K; kk += 32) {
    __syncthreads();
    // Warm GL2 two chunks ahead (global_prefetch_b8), stage next chunk now.
    if (kk + 64 < K) {
      __builtin_prefetch(&A[(size_t)(tileM + tid) * lda + kk + 64], 0, 0);
      if (!TRANSB) {
        if (tid < 32) __builtin_prefetch(&B[(size_t)(kk + 64 + tid) * ldb + tileN], 0, 0);
      } else {
        if (tid < 64) __builtin_prefetch(&B[(size_t)(tileN + tid) * ldb + kk + 64], 0, 0);
      }
    }
    if (kk + 32 < K) stage(cur ^ 1, kk + 32);

    const int aoff = cur * BUF_FLOATS;
    const int boff = cur * BUF_FLOATS + AS_FLOATS;
    const int abase = (wave << 5) + l15;
#pragma unroll
    for (int k0 = 0; k0 < 32; k0 += 4) {
      int koff = k0 + (lhalf << 1);
      v2f a0, a1;
      a0.x = smem[aoff + abase * AS_STRIDE + koff];
      a0.y = smem[aoff + abase * AS_STRIDE + koff + 1];
      a1.x = smem[aoff + (abase + 16) * AS_STRIDE + koff];
      a1.y = smem[aoff + (abase + 16) * AS_STRIDE + koff + 1];
#pragma unroll
      for (int nt = 0; nt < 4; ++nt) {
        v2f b;
        int bc = (nt << 4) + l15;
        b.x = smem[boff + koff * BS_STRIDE + bc];
        b.y = smem[boff + (koff + 1) * BS_STRIDE + bc];
        acc[0][nt] = __builtin_amdgcn_wmma_f32_16x16x4_f32(
            false, a0, false, b, (short)0, acc[0][nt], false, false);
        acc[1][nt] = __builtin_amdgcn_wmma_f32_16x16x4_f32(
            false, a1, false, b, (short)0, acc[1][nt], false, false);
      }
    }
    cur ^= 1;
  }
}

// Map accumulator element (rt, nt, r) to (row, col) inside the 128x64 tile.
#define ACC_ROW(wave, lhalf, rt, r)  (((wave) << 5) + ((rt) << 4) + (r) + ((lhalf) << 3))
#define ACC_COL(l15, nt)             (((nt) << 4) + (l15))

// ------------------------------------------------------------------
// LayerNorm: one block per row of x [L x D]
// ------------------------------------------------------------------
__global__ void ln_kernel(const float* __restrict__ x, const float* __restrict__ gamma,
                          const float* __restrict__ beta, float* __restrict__ h) {
  __shared__ float s1[256], s2[256];
  int row = blockIdx.x;
  const float* xr = x + (size_t)row * DDIM;
  float sum = 0.f, sumsq = 0.f;
  for (int i = threadIdx.x; i < DDIM; i += 256) { float v = xr[i]; sum += v; sumsq += v * v; }
  s1[threadIdx.x] = sum; s2[threadIdx.x] = sumsq; __syncthreads();
  for (int s = 128; s > 0; s >>= 1) {
    if (threadIdx.x < s) { s1[threadIdx.x] += s1[threadIdx.x + s]; s2[threadIdx.x] += s2[threadIdx.x + s]; }
    __syncthreads();
  }
  float mean = s1[0] * (1.f / DDIM);
  float var  = s2[0] * (1.f / DDIM) - mean * mean;
  float rstd = rsqrtf(var + 1e-5f);
  for (int i = threadIdx.x; i < DDIM; i += 256)
    h[(size_t)row * DDIM + i] = (xr[i] - mean) * rstd * gamma[i] + beta[i];
}

// ------------------------------------------------------------------
// QKV projection: out head-major [H][L][DH]
// ------------------------------------------------------------------
__global__ void gemm_qkv_kernel(const float* __restrict__ h, const float* __restrict__ W,
                                float* __restrict__ out_hm) {
  int tileN = blockIdx.x * 64, tileM = blockIdx.y * 128;
  v8f acc[2][4] = {};
  gemm_tile_128x64<false>(h, W, DDIM, DDIM, DDIM, tileM, tileN, acc);
  int wave = threadIdx.x >> 5, lane = threadIdx.x & 31;
  int l15 = lane & 15, lh = lane >> 4;
#pragma unroll
  for (int rt = 0; rt < 2; ++rt)
#pragma unroll
    for (int nt = 0; nt < 4; ++nt)
#pragma unroll
      for (int r = 0; r < 8; ++r) {
        int m = tileM + ACC_ROW(wave, lh, rt, r);
        int n = tileN + ACC_COL(l15, nt);
        int head = n >> 6, dh = n & 63;
        out_hm[((size_t)head * LL + m) * DHD + dh] = acc[rt][nt][r];
      }
}

// ------------------------------------------------------------------
// RoPE in-place on head-major tensor [H][L][DH], first 16 dims.
// ------------------------------------------------------------------
__global__ void rope_kernel(float* __restrict__ t) {
  int idx = blockIdx.x * blockDim.x + threadIdx.x;   // over H*L
  if (idx >= NH * LL) return;
  int l = idx & (LL - 1);
  float* p = t + (size_t)idx * DHD;
  float xr[16];
#pragma unroll
  for (int d = 0; d < 16; ++d) xr[d] = p[d];
#pragma unroll
  for (int d = 0; d < 16; ++d) {
    int fi = d & 7;
    // inv_freq[i] = 10000^(-i/8) = exp(-i * ln(10000)/8)
    float invf = __expf(-(float)fi * (9.2103403720f / 8.0f));
    float f = (float)l * invf;
    float c = __cosf(f), s = __sinf(f);
    float rot = (d < 8) ? -xr[d + 8] : xr[d - 8];
    p[d] = xr[d] * c + rot * s;
  }
}

// ------------------------------------------------------------------
// Scores: dots = q*k^T*scale + prev + alibi_causal  (per head)
// ------------------------------------------------------------------
__global__ void gemm_scores_kernel(const float* __restrict__ q, const float* __restrict__ k,
                                   const float* __restrict__ prev, float* __restrict__ dots) {
  int head = blockIdx.z;
  const float* qh = q + (size_t)head * LL * DHD;
  const float* kh = k + (size_t)head * LL * DHD;
  int tileN = blockIdx.x * 64, tileM = blockIdx.y * 128;
  v8f acc[2][4] = {};
  gemm_tile_128x64<true>(qh, kh, DHD, DHD, DHD, tileM, tileN, acc);
  float slope = exp2f(-0.5f * (float)(head + 1));   // 2^(-8*(h+1)/H), H=16
  const float* prevh = prev + (size_t)head * LL * LL;
  float* dsth = dots + (size_t)head * LL * LL;
  int wave = threadIdx.x >> 5, lane = threadIdx.x & 31;
  int l15 = lane & 15, lh = lane >> 4;
#pragma unroll
  for (int rt = 0; rt < 2; ++rt)
#pragma unroll
    for (int nt = 0; nt < 4; ++nt)
#pragma unroll
      for (int r = 0; r < 8; ++r) {
        int m = tileM + ACC_ROW(wave, lh, rt, r);
        int n = tileN + ACC_COL(l15, nt);
        int rel = m - n;
        float bias = (rel >= 0) ? (-slope * (float)rel) : NEGV;
        dsth[(size_t)m * LL + n] = acc[rt][nt][r] * SCALE_ + prevh[(size_t)m * LL + n] + bias;
      }
}

// ------------------------------------------------------------------
// Row softmax over last axis of dots [H*L][L] -> attn (float4 streamed)
// ------------------------------------------------------------------
__global__ void softmax_kernel(const float* __restrict__ dots, float* __restrict__ attn) {
  __shared__ float red[256];
  size_t row = blockIdx.x;
  const float4* src4 = (const float4*)(dots + row * LL);
  float4* dst4 = (float4*)(attn + row * LL);
  float mx = -INFINITY;
  for (int i = threadIdx.x; i < LL / 4; i += 256) {
    float4 v = src4[i];
    mx = fmaxf(mx, fmaxf(fmaxf(v.x, v.y), fmaxf(v.z, v.w)));
  }
  red[threadIdx.x] = mx; __syncthreads();
  for (int s = 128; s > 0; s >>= 1) {
    if (threadIdx.x < s) red[threadIdx.x] = fmaxf(red[threadIdx.x], red[threadIdx.x + s]);
    __syncthreads();
  }
  mx = red[0]; __syncthreads();
  float sum = 0.f;
  for (int i = threadIdx.x; i < LL / 4; i += 256) {
    float4 v = src4[i];
    sum += __expf(v.x - mx) + __expf(v.y - mx) + __expf(v.z - mx) + __expf(v.w - mx);
  }
  red[threadIdx.x] = sum; __syncthreads();
  for (int s = 128; s > 0; s >>= 1) {
    if (threadIdx.x < s) red[threadIdx.x] += red[threadIdx.x + s];
    __syncthreads();
  }
  float inv = 1.f / red[0];
  for (int i = threadIdx.x; i < LL / 4; i += 256) {
    float4 v = src4[i];
    float4 o;
    o.x = __expf(v.x - mx) * inv;
    o.y = __expf(v.y - mx) * inv;
    o.z = __expf(v.z - mx) * inv;
    o.w = __expf(v.w - mx) * inv;
    dst4[i] = o;
  }
}

// ------------------------------------------------------------------
// AV: o[l][head*64+dh] = sum_j attn[head][l][j] * v[head][j][dh]
// ------------------------------------------------------------------
__global__ void gemm_av_kernel(const float* __restrict__ attn, const float* __restrict__ v,
                               float* __restrict__ o) {
  int head = blockIdx.z;
  const float* ah = attn + (size_t)head * LL * LL;
  const float* vh = v + (size_t)head * LL * DHD;
  int tileM = blockIdx.y * 128;            // tileN == 0 (N = 64)
  v8f acc[2][4] = {};
  gemm_tile_128x64<false>(ah, vh, LL, DHD, LL, tileM, 0, acc);
  int wave = threadIdx.x >> 5, lane = threadIdx.x & 31;
  int l15 = lane & 15, lh = lane >> 4;
#pragma unroll
  for (int rt = 0; rt < 2; ++rt)
#pragma unroll
    for (int nt = 0; nt < 4; ++nt)
#pragma unroll
      for (int r = 0; r < 8; ++r) {
        int m = tileM + ACC_ROW(wave, lh, rt, r);
        int n = ACC_COL(l15, nt);          // 0..63
        o[(size_t)m * DDIM + head * DHD + n] = acc[rt][nt][r];
      }
}

// ------------------------------------------------------------------
// Plain GEMM [L x D] x [D x D] -> [L x D]
// ------------------------------------------------------------------
__global__ void gemm_plain_kernel(const float* __restrict__ A, const float* __restrict__ W,
                                  float* __restrict__ C) {
  int tileN = blockIdx.x * 64, tileM = blockIdx.y * 128;
  v8f acc[2][4] = {};
  gemm_tile_128x64<false>(A, W, DDIM, DDIM, DDIM, tileM, tileN, acc);
  int wave = threadIdx.x >> 5, lane = threadIdx.x & 31;
  int l15 = lane & 15, lh = lane >> 4;
#pragma unroll
  for (int rt = 0; rt < 2; ++rt)
#pragma unroll
    for (int nt = 0; nt < 4; ++nt)
#pragma unroll
      for (int r = 0; r < 8; ++r) {
        int m = tileM + ACC_ROW(wave, lh, rt, r);
        int n = tileN + ACC_COL(l15, nt);
        C[(size_t)m * DDIM + n] = acc[rt][nt][r];
      }
}

// ------------------------------------------------------------------
// y = x + sigmoid(gg + bg) * (og + bo)   (float4 streamed)
// ------------------------------------------------------------------
__global__ void finalize_kernel(const float* __restrict__ x, const float* __restrict__ og,
                                const float* __restrict__ gg, const float* __restrict__ bo,
                                const float* __restrict__ bg, float* __restrict__ y) {
  int idx = blockIdx.x * 256 + threadIdx.x;          // over (L*D)/4
  if (idx >= (LL * DDIM) / 4) return;
  int col4 = idx & ((DDIM / 4) - 1);
  float4 xv = ((const float4*)x)[idx];
  float4 ov = ((const float4*)og)[idx];
  float4 gv = ((const float4*)gg)[idx];
  float4 bov = ((const float4*)bo)[col4];
  float4 bgv = ((const float4*)bg)[col4];
  float4 o;
  o.x = xv.x + (1.f / (1.f + __expf(-(gv.x + bgv.x)))) * (ov.x + bov.x);
  o.y = xv.y + (1.f / (1.f + __expf(-(gv.y + bgv.y)))) * (ov.y + bov.y);
  o.z = xv.z + (1.f / (1.f + __expf(-(gv.z + bgv.z)))) * (ov.z + bov.z);
  o.w = xv.w + (1.f / (1.f + __expf(-(gv.w + bgv.w)))) * (ov.w + bov.w);
  ((float4*)y)[idx] = o;
}

// ------------------------------------------------------------------
extern "C" void kernel_launch(void* const* d_in, const int* in_sizes, int n_in,
                              void* d_out, int out_size, void* d_ws, size_t ws_size,
                              hipStream_t stream) {
  const float* x     = (const float*)d_in[0];
  const float* prev  = (const float*)d_in[1];
  const float* Wq    = (const float*)d_in[2];
  const float* Wk    = (const float*)d_in[3];
  const float* Wv    = (const float*)d_in[4];
  const float* Wo    = (const float*)d_in[5];
  const float* bo    = (const float*)d_in[6];
  const float* gamma = (const float*)d_in[7];
  const float* beta  = (const float*)d_in[8];
  const float* Wg    = (const float*)d_in[9];
  const float* bg    = (const float*)d_in[10];

  float* ws = (float*)d_ws;
  const size_t LD = (size_t)LL * DDIM;   // 2,097,152
  float* h  = ws;
  float* q  = ws + 1 * LD;
  float* k  = ws + 2 * LD;
  float* v  = ws + 3 * LD;
  float* o  = ws + 4 * LD;
  float* og = ws + 5 * LD;
  float* gg = ws + 6 * LD;

  float* y    = (float*)d_out;
  float* attn = y + LD;                         // (B,H,L,L)
  float* dots = attn + (size_t)NH * LL * LL;    // (B,H,L,L)

  ln_kernel<<<LL, 256, 0, stream>>>(x, gamma, beta, h);

  dim3 gP(DDIM / 64, LL / 128);
  gemm_qkv_kernel<<<gP, 128, 0, stream>>>(h, Wq, q);
  gemm_qkv_kernel<<<gP, 128, 0, stream>>>(h, Wk, k);
  gemm_qkv_kernel<<<gP, 128, 0, stream>>>(h, Wv, v);

  int nrope = NH * LL;
  rope_kernel<<<(nrope + 127) / 128, 128, 0, stream>>>(q);
  rope_kernel<<<(nrope + 127) / 128, 128, 0, stream>>>(k);
  rope_kernel<<<(nrope + 127) / 128, 128, 0, stream>>>(v);

  dim3 gS(LL / 64, LL / 128, NH);
  gemm_scores_kernel<<<gS, 128, 0, stream>>>(q, k, prev, dots);

  softmax_kernel<<<NH * LL, 256, 0, stream>>>(dots, attn);

  dim3 gA(1, LL / 128, NH);
  gemm_av_kernel<<<gA, 128, 0, stream>>>(attn, v, o);

  gemm_plain_kernel<<<gP, 128, 0, stream>>>(o, Wo, og);
  gemm_plain_kernel<<<gP, 128, 0, stream>>>(x, Wg, gg);

  finalize_kernel<<<((LL * DDIM) / 4 + 255) / 256, 256, 0, stream>>>(x, og, gg, bo, bg, y);
}